// NodeCentric_2482491097663
// MI455X (gfx1250) — compile-verified
//
#include <hip/hip_runtime.h>

typedef float v2f __attribute__((ext_vector_type(2)));
typedef float v8f __attribute__((ext_vector_type(8)));

#define NNODES 100000
#define NEDGES 1600000
#define IN_X   128
#define IN_E   32
#define OUT_X  128
#define OUT_TOT 256

// ---------------------------------------------------------------------------
// Kernel 1: zero the aggregation workspace (d_ws is poisoned 0xAA by harness).
// float4 stores; 3.2M floats = 800K float4.
// ---------------------------------------------------------------------------
__global__ void zero_agg_kernel(float4* __restrict__ agg4) {
    int i = blockIdx.x * blockDim.x + threadIdx.x;
    if (i < (NNODES * IN_E) / 4) agg4[i] = make_float4(0.f, 0.f, 0.f, 0.f);
}

// ---------------------------------------------------------------------------
// Kernel 2: scatter-add edge_attr[E,32] into agg[N,32] keyed by edge_index[0].
// One thread per (edge, feature); lane index == feature, so the 32 atomics of
// one edge hit 32 consecutive floats and edge_attr reads are fully coalesced.
// agg (12.8 MB) is L2-resident (192 MB L2) -> atomics resolve on-chip.
// ---------------------------------------------------------------------------
__global__ void scatter_edge_kernel(const float* __restrict__ edge_attr,
                                    const long long* __restrict__ src_idx,
                                    float* __restrict__ agg) {
    long long gid = (long long)blockIdx.x * blockDim.x + threadIdx.x;
    if (gid >= (long long)NEDGES * IN_E) return;
    int e = (int)(gid >> 5);   // edge id
    int f = (int)(gid & 31);   // feature id
    long long s = src_idx[e];  // edge_index row 0 (int64)
    atomicAdd(&agg[s * IN_E + f], edge_attr[gid]);
}

// ---------------------------------------------------------------------------
// One 16x16 f32 output tile via V_WMMA_F32_16X16X4_F32 with compile-time K so
// the K-loop fully unrolls (no exec-masked loop, WMMAs pipeline with loads).
//
// Fragment layouts (ISA 7.12.2, f32):
//   A 16x4 : lane l<16 -> {A[m][k], A[m][k+1]}, l>=16 -> {A[m][k+2], A[m][k+3]}
//   B 4x16 : lane l<16 -> {B[k][n], B[k+1][n]}, l>=16 -> {B[k+2][n], B[k+3][n]}
//   C/D    : VGPR r -> row (r + 8*half), col n = lane&15
// ---------------------------------------------------------------------------
template <int K>
static __device__ __forceinline__ v8f tile_gemm(const float* __restrict__ Arow,
                                                const float* __restrict__ Wcol,
                                                int half) {
    // Arow = &A[(rowbase + m) * K]           (m = lane & 15)
    // Wcol = &W[col + n]                     (row stride 128 floats)
    v8f acc = {};
#pragma unroll
    for (int k = 0; k < K; k += 4) {
        const int ka = k + 2 * half;
        v2f a, b;
        a.x = Arow[ka];
        a.y = Arow[ka + 1];
        b.x = Wcol[ka * 128];
        b.y = Wcol[(ka + 1) * 128];
        // 8 args: (neg_a, A, neg_b, B, c_mod, C, reuse_a, reuse_b)
        acc = __builtin_amdgcn_wmma_f32_16x16x4_f32(
                  false, a, false, b, (short)0, acc, false, false);
    }
    return acc;
}

// ---------------------------------------------------------------------------
// Kernel 3: fused  out[:, 0:128]   = x   @ Wx + bx   (K = 128)
//                  out[:, 128:256] = agg @ We + be   (K = 32)
// Block = 512 threads = 16 waves; each wave owns one 16x16 tile of the 16x256
// fused output strip. Wave-uniform branch picks the compile-time-K path, so
// EXEC is all-ones at every WMMA. N = 100000 = 6250 * 16 exactly.
// ---------------------------------------------------------------------------
__global__ void fused_gemm_kernel(const float* __restrict__ x,
                                  const float* __restrict__ agg,
                                  const float* __restrict__ Wx,
                                  const float* __restrict__ bx,
                                  const float* __restrict__ We,
                                  const float* __restrict__ be,
                                  float* __restrict__ out) {
    const int rowbase = blockIdx.x * 16;
    const int wave = threadIdx.x >> 5;   // 0..15 (uniform per wave)
    const int lane = threadIdx.x & 31;
    const int n    = lane & 15;          // output column within tile
    const int half = lane >> 4;          // 0 or 1
    const int m    = n;                  // A row within the 16-row strip
    const int col  = (wave & 7) * 16;    // column within the 128-wide W matrix

    v8f acc;
    int outcol;
    float bv;
    if (wave < 8) {
        acc = tile_gemm<IN_X>(x + (long long)(rowbase + m) * IN_X, Wx + col + n, half);
        bv = bx[col + n];
        outcol = col;
    } else {
        acc = tile_gemm<IN_E>(agg + (long long)(rowbase + m) * IN_E, We + col + n, half);
        bv = be[col + n];
        outcol = OUT_X + col;
    }

#pragma unroll
    for (int r = 0; r < 8; ++r) {
        const int row = rowbase + r + 8 * half;
        out[(long long)row * OUT_TOT + outcol + n] = acc[r] + bv;
    }
}

// ---------------------------------------------------------------------------
extern "C" void kernel_launch(void* const* d_in, const int* in_sizes, int n_in,
                              void* d_out, int out_size, void* d_ws, size_t ws_size,
                              hipStream_t stream) {
    const float*     x          = (const float*)d_in[0];
    const long long* edge_index = (const long long*)d_in[1]; // [2,E] int64; row 0 = first E
    const float*     edge_attr  = (const float*)d_in[2];
    const float*     Wx         = (const float*)d_in[3];
    const float*     bx         = (const float*)d_in[4];
    const float*     We         = (const float*)d_in[5];
    const float*     be         = (const float*)d_in[6];
    float*           out        = (float*)d_out;
    float*           agg        = (float*)d_ws;             // [N, 32] f32 = 12.8 MB

    (void)in_sizes; (void)n_in; (void)out_size; (void)ws_size;

    {
        int total4 = (NNODES * IN_E) / 4;
        zero_agg_kernel<<<(total4 + 255) / 256, 256, 0, stream>>>((float4*)agg);
    }
    {
        long long total = (long long)NEDGES * IN_E;          // 51.2M
        int blocks = (int)((total + 255) / 256);
        scatter_edge_kernel<<<blocks, 256, 0, stream>>>(edge_attr, edge_index, agg);
    }
    {
        fused_gemm_kernel<<<NNODES / 16, 512, 0, stream>>>(x, agg, Wx, bx, We, be, out);
    }
}